// GAT_10995116277936
// MI455X (gfx1250) — compile-verified
//
#include <hip/hip_runtime.h>
#include <math.h>

// ---------------------------------------------------------------------------
// GAT 3-layer forward for MI455X (gfx1250, wave32).
// GEMMs use V_WMMA_F32_16X16X4_F32 (full fp32 precision; GEMM is <5% of the
// roofline — edge gather/scatter (~2.8 GB) dominates at 23.3 TB/s HBM).
// Each wave computes a 16x64 output strip: one A fragment feeds 4 WMMAs/K-step.
// ---------------------------------------------------------------------------

typedef __attribute__((ext_vector_type(2))) float v2f;
typedef __attribute__((ext_vector_type(8))) float v8f;

#define WMMA_F32(A, B, C)                                                     \
    __builtin_amdgcn_wmma_f32_16x16x4_f32(false, (A), false, (B), (short)0,   \
                                          (C), false, false)

// ---------------- GEMM: F[N,D] = act(X[N,K]) @ W[K,D] ----------------------
// fp32 WMMA operand layout (ISA 7.12.2):
//   A (16x4):  lanes 0-15 row M=lane   {K=k0,k0+1}; lanes 16-31 {K=k0+2,k0+3}
//   B (4x16):  mirrors A with M<->N (row of B striped across lanes per VGPR)
//   C (16x16): VGPR r -> row r (lanes 0-15) / row 8+r (lanes 16-31), col=lane&15
template <int RELU>
__global__ __launch_bounds__(32) void gemm_wmma_f32(
    const float* __restrict__ X, const float* __restrict__ W,
    float* __restrict__ F, int K, int D)
{
    const int lane = threadIdx.x;
    const int half = lane >> 4;          // 0 or 1
    const int l    = lane & 15;
    const int m0   = blockIdx.x * 16;
    const int n0   = blockIdx.y * 64;    // 4 n-tiles of 16 per wave

    const float* __restrict__ xrow = X + (size_t)(m0 + l) * K;
    v8f c0 = {}, c1 = {}, c2 = {}, c3 = {};

    for (int k0 = 0; k0 < K; k0 += 4) {
        if ((k0 & 31) == 0 && k0 + 32 < K)
            __builtin_prefetch(xrow + k0 + 32, 0, 1);   // global_prefetch_b8

        float a0 = xrow[k0 + 2 * half];
        float a1 = xrow[k0 + 2 * half + 1];
        if (RELU) { a0 = (a0 > 0.0f) ? a0 : 0.0f; a1 = (a1 > 0.0f) ? a1 : 0.0f; }
        v2f a; a[0] = a0; a[1] = a1;

        const float* wr0 = W + (size_t)(k0 + 2 * half) * D + n0 + l;
        const float* wr1 = wr0 + D;
        v2f b0, b1, b2, b3;
        b0[0] = wr0[0];  b0[1] = wr1[0];
        b1[0] = wr0[16]; b1[1] = wr1[16];
        b2[0] = wr0[32]; b2[1] = wr1[32];
        b3[0] = wr0[48]; b3[1] = wr1[48];

        c0 = WMMA_F32(a, b0, c0);
        c1 = WMMA_F32(a, b1, c1);
        c2 = WMMA_F32(a, b2, c2);
        c3 = WMMA_F32(a, b3, c3);
    }

    float* out = F + (size_t)(m0 + half * 8) * D + n0 + l;
#pragma unroll
    for (int r = 0; r < 8; ++r) {
        out[(size_t)r * D +  0] = c0[r];
        out[(size_t)r * D + 16] = c1[r];
        out[(size_t)r * D + 32] = c2[r];
        out[(size_t)r * D + 48] = c3[r];
    }
}

// ---------------- el = F@al, er = F@ar (one wave per node) -----------------
__global__ __launch_bounds__(32) void node_dots(
    const float* __restrict__ F, const float* __restrict__ al,
    const float* __restrict__ ar, float* __restrict__ el,
    float* __restrict__ er, int D)
{
    const int n = blockIdx.x;
    const int lane = threadIdx.x;
    float sl = 0.0f, sr = 0.0f;
    const float* row = F + (size_t)n * D;
    for (int j = lane; j < D; j += 32) {
        float f = row[j];
        sl = fmaf(f, al[j], sl);
        sr = fmaf(f, ar[j], sr);
    }
#pragma unroll
    for (int o = 16; o > 0; o >>= 1) {
        sl += __shfl_xor(sl, o, 32);
        sr += __shfl_xor(sr, o, 32);
    }
    if (lane == 0) { el[n] = sl; er[n] = sr; }
}

// ---------------- init m = -inf, z = 0 -------------------------------------
__global__ void init_mz(float* __restrict__ m, float* __restrict__ z, int N)
{
    int i = blockIdx.x * blockDim.x + threadIdx.x;
    if (i < N) { m[i] = -INFINITY; z[i] = 0.0f; }
}

// Order-independent float atomic max (sign-split onto native int atomics).
__device__ inline void atomicMaxF(float* addr, float val)
{
    unsigned int u = __float_as_uint(val);
    if (!(u & 0x80000000u))
        atomicMax((int*)addr, (int)u);            // non-negative: int max
    else
        atomicMin((unsigned int*)addr, u);        // negative: unsigned min
}

// ---------------- edge pass 1: leaky-relu score + segment max --------------
__global__ void edge_score_max(
    const float* __restrict__ el, const float* __restrict__ er,
    const int* __restrict__ src, const int* __restrict__ dst,
    float* __restrict__ score, float* __restrict__ m, int E)
{
    int e = blockIdx.x * blockDim.x + threadIdx.x;
    if (e >= E) return;
    float s = el[src[e]] + er[dst[e]];
    s = (s > 0.0f) ? s : 0.2f * s;                // leaky_relu(0.2)
    score[e] = s;
    atomicMaxF(&m[dst[e]], s);
}

// ---------------- edge pass 2: exp(s - m[dst]) + segment sum ---------------
__global__ void edge_exp_sum(
    float* __restrict__ score, const float* __restrict__ m,
    const int* __restrict__ dst, float* __restrict__ z, int E)
{
    int e = blockIdx.x * blockDim.x + threadIdx.x;
    if (e >= E) return;
    int d = dst[e];
    float ex = expf(score[e] - m[d]);
    score[e] = ex;                                // in place: score -> ex
    atomicAdd(&z[d], ex);
}

// ---------------- edge pass 3: agg[dst] += (ex/z[dst]) * F[src] ------------
// threads_per_edge = D/4 lanes; float4 gather, 4 scalar f32 atomic adds.
__global__ void edge_aggregate(
    const float* __restrict__ F, const float* __restrict__ ex,
    const float* __restrict__ z, const int* __restrict__ src,
    const int* __restrict__ dst, float* __restrict__ agg, int E, int D)
{
    const int tpe = D >> 2;                       // 32 (D=128) or 16 (D=64)
    const int epb = blockDim.x / tpe;
    const int e = blockIdx.x * epb + (int)threadIdx.x / tpe;
    if (e >= E) return;
    const int t = (int)threadIdx.x % tpe;
    const int s = src[e], d = dst[e];
    const float a = ex[e] / fmaxf(z[d], 1e-9f);
    const float4 v = *((const float4*)(F + (size_t)s * D) + t);
    float* o = agg + (size_t)d * D + 4 * t;
    atomicAdd(o + 0, a * v.x);
    atomicAdd(o + 1, a * v.y);
    atomicAdd(o + 2, a * v.z);
    atomicAdd(o + 3, a * v.w);
}

// ---------------- log_softmax over D=64, one wave per node -----------------
__global__ __launch_bounds__(32) void log_softmax64(
    const float* __restrict__ X, float* __restrict__ out, int N)
{
    const int n = blockIdx.x;
    const int lane = threadIdx.x;
    float2 v = *((const float2*)(X + (size_t)n * 64) + lane);
    float mx = fmaxf(v.x, v.y);
#pragma unroll
    for (int o = 16; o > 0; o >>= 1) mx = fmaxf(mx, __shfl_xor(mx, o, 32));
    float s = expf(v.x - mx) + expf(v.y - mx);
#pragma unroll
    for (int o = 16; o > 0; o >>= 1) s += __shfl_xor(s, o, 32);
    const float lg = mx + logf(s);
    float2 r; r.x = v.x - lg; r.y = v.y - lg;
    *((float2*)(out + (size_t)n * 64) + lane) = r;
}

// ---------------------------------------------------------------------------
extern "C" void kernel_launch(void* const* d_in, const int* in_sizes, int n_in,
                              void* d_out, int out_size, void* d_ws, size_t ws_size,
                              hipStream_t stream)
{
    // inputs: h, src, dst, W1, al1, ar1, W2, al2, ar2, W3, al3, ar3
    const float* h   = (const float*)d_in[0];
    const int*   src = (const int*)  d_in[1];
    const int*   dst = (const int*)  d_in[2];
    const float* W1  = (const float*)d_in[3];
    const float* al1 = (const float*)d_in[4];
    const float* ar1 = (const float*)d_in[5];
    const float* W2  = (const float*)d_in[6];
    const float* al2 = (const float*)d_in[7];
    const float* ar2 = (const float*)d_in[8];
    const float* W3  = (const float*)d_in[9];
    const float* al3 = (const float*)d_in[10];
    const float* ar3 = (const float*)d_in[11];

    const int DH   = in_sizes[4];             // 128
    const int DOUT = in_sizes[10];            // 64
    const int DIN  = in_sizes[3] / DH;        // 256
    const int N    = in_sizes[0] / DIN;       // 50000 (multiple of 16)
    const int E    = in_sizes[1];             // 800000

    // workspace layout (floats)
    float* ws    = (float*)d_ws;
    float* f_buf = ws;                                  // N*128
    float* agg   = f_buf + (size_t)N * 128;             // N*128
    float* el    = agg   + (size_t)N * 128;             // N
    float* er    = el + N;                              // N
    float* m     = er + N;                              // N
    float* z     = m  + N;                              // N
    float* score = z  + N;                              // E

    const int TB = 256;

    auto run_layer = [&](const float* X, const float* W, const float* al,
                         const float* ar, int K, int D, int relu_in,
                         float* f, float* ag) {
        dim3 gg(N / 16, D / 64);              // 16x64 strip per wave
        if (relu_in)
            gemm_wmma_f32<1><<<gg, 32, 0, stream>>>(X, W, f, K, D);
        else
            gemm_wmma_f32<0><<<gg, 32, 0, stream>>>(X, W, f, K, D);
        node_dots<<<N, 32, 0, stream>>>(f, al, ar, el, er, D);
        init_mz<<<(N + TB - 1) / TB, TB, 0, stream>>>(m, z, N);
        hipMemsetAsync(ag, 0, (size_t)N * D * sizeof(float), stream);
        edge_score_max<<<(E + TB - 1) / TB, TB, 0, stream>>>(el, er, src, dst,
                                                             score, m, E);
        edge_exp_sum<<<(E + TB - 1) / TB, TB, 0, stream>>>(score, m, dst, z, E);
        const int epb = TB / (D >> 2);
        edge_aggregate<<<(E + epb - 1) / epb, TB, 0, stream>>>(f, score, z, src,
                                                               dst, ag, E, D);
    };

    // layer 1: h -> agg (x1);  layer 2: relu(x1) -> agg (x2, in place: GEMM
    // consumes agg before the memset, stream order guarantees safety);
    // layer 3: relu(x2) -> agg (logits, N x 64)
    run_layer(h,   W1, al1, ar1, DIN, DH,   0, f_buf, agg);
    run_layer(agg, W2, al2, ar2, DH,  DH,   1, f_buf, agg);
    run_layer(agg, W3, al3, ar3, DH,  DOUT, 1, f_buf, agg);

    log_softmax64<<<N, 32, 0, stream>>>(agg, (float*)d_out, N);
    (void)n_in; (void)out_size; (void)ws_size;
}